// CustomModel_13657996001613
// MI455X (gfx1250) — compile-verified
//
#include <hip/hip_runtime.h>
#include <hip/hip_bf16.h>

// ---------------------------------------------------------------------------
// CDNA5 fp32 WMMA types
// ---------------------------------------------------------------------------
typedef float v2f __attribute__((ext_vector_type(2)));
typedef float v8f __attribute__((ext_vector_type(8)));

#define N_NODES 1024
#define N_EDGES 32768

// ---------------------------------------------------------------------------
// Generic fp32 WMMA GEMM:  C[M,N] = op(A[M,K] @ B[K,N] (+bias) (relu))
// One wave -> 16 x (16*NT) output tile (NT accumulators share one A frag:
// NT independent WMMA dependency chains, A reuse x NT).
// Requires: M % 64 == 0 (4 waves/block stack in M), N % (16*NT) == 0, K % 4 == 0.
// fuse: bit0 = add bias (len N), bit1 = relu.
// Addressing uses uniform base + 32-bit per-lane element offsets so loads take
// the saddr+voffset global-load form (minimal per-iteration VALU).
// ---------------------------------------------------------------------------
template <int NT>
__global__ __launch_bounds__(128)
void gemm_f32_wmma(const float* __restrict__ A, const float* __restrict__ B,
                   const float* __restrict__ bias, float* __restrict__ C,
                   int M, int N, int K, int fuse)
{
    const int lane = threadIdx.x & 31;
    const int wave = threadIdx.x >> 5;
    const int tileN = blockIdx.x * (16 * NT);
    const int tileM = (blockIdx.y * 4 + wave) * 16;

    const int half = lane >> 4;   // selects K-pair {0,1} vs {2,3} within k0..k0+3
    const int lmod = lane & 15;   // row (A) / column (B,D) within tile

    v8f acc[NT];
    #pragma unroll
    for (int t = 0; t < NT; ++t) { v8f z = {}; acc[t] = z; }

    // 32-bit element offsets from uniform bases.
    int aoff = (tileM + lmod) * K + 2 * half;   // A[m][k0 + 2*half + {0,1}]
    int boff = 2 * half * N + tileN + lmod;     // B[k0 + 2*half + {0,1}][n (+16t)]

    for (int k0 = 0; k0 < K; k0 += 4) {
        v2f a;
        a.x = A[aoff];
        a.y = A[aoff + 1];

        // Unconditional speculative prefetches (silently dropped if OOB).
        __builtin_prefetch(A + aoff + 64, 0, 1);
        __builtin_prefetch(B + boff + 16 * N, 0, 1);

        #pragma unroll
        for (int t = 0; t < NT; ++t) {
            v2f b;
            b.x = B[boff + 16 * t];
            b.y = B[boff + N + 16 * t];
            acc[t] = __builtin_amdgcn_wmma_f32_16x16x4_f32(false, a, false, b,
                                                           (short)0, acc[t],
                                                           false, false);
        }
        aoff += 4;
        boff += 4 * N;
    }

    #pragma unroll
    for (int t = 0; t < NT; ++t) {
        float bs = 0.f;
        if (fuse & 1) bs = bias[tileN + 16 * t + lmod];
        #pragma unroll
        for (int v = 0; v < 8; ++v) {
            const int row = tileM + v + 8 * half;   // D: VGPR v -> row v (+8 hi half)
            float x = acc[t][v] + bs;
            if (fuse & 2) x = fmaxf(x, 0.f);
            C[(size_t)row * N + tileN + 16 * t + lmod] = x;
        }
    }
}

// ---------------------------------------------------------------------------
// Tiny output heads + argmax routing.
// logits = H2A @ mlp_W3 + mlp_b3   [N,3];  r = soft one-hot(argmax)
// g      = H2B @ m1_W3  + m1_b3    [N,2]
// ---------------------------------------------------------------------------
__global__ void route_kernel(const float* __restrict__ h2a, const float* __restrict__ h2b,
                             const float* __restrict__ W3a, const float* __restrict__ b3a,
                             const float* __restrict__ W3b, const float* __restrict__ b3b,
                             float* __restrict__ r, float* __restrict__ g)
{
    int n = blockIdx.x * blockDim.x + threadIdx.x;
    if (n >= N_NODES) return;

    float l0 = b3a[0], l1 = b3a[1], l2 = b3a[2];
    const float* h = h2a + (size_t)n * 512;
    for (int i = 0; i < 512; ++i) {
        float hv = h[i];
        l0 += hv * W3a[i * 3 + 0];
        l1 += hv * W3a[i * 3 + 1];
        l2 += hv * W3a[i * 3 + 2];
    }
    int am = 0; float mx = l0;               // first-max semantics like jnp.argmax
    if (l1 > mx) { mx = l1; am = 1; }
    if (l2 > mx) { mx = l2; am = 2; }
    const float TAU = 0.7f, LO = (1.0f - 0.7f) * 0.5f;
    r[n * 3 + 0] = (am == 0) ? TAU : LO;
    r[n * 3 + 1] = (am == 1) ? TAU : LO;
    r[n * 3 + 2] = (am == 2) ? TAU : LO;

    float g0 = b3b[0], g1 = b3b[1];
    const float* hb = h2b + (size_t)n * 512;
    for (int i = 0; i < 512; ++i) {
        float hv = hb[i];
        g0 += hv * W3b[i * 2 + 0];
        g1 += hv * W3b[i * 2 + 1];
    }
    g[n * 2 + 0] = g0;
    g[n * 2 + 1] = g1;
}

// ---------------------------------------------------------------------------
// Collapse the last axis of the 3D einsum weights:
//   Cs[j,k]  = sum_l class2[j,k,l]  (2x768, l<64)
//   Ws1[j,k] = sum_l wm1[j,k,l]     (3x768, l<256)
//   Ws2[j,k] = sum_l wm2[j,k,l]     (3x256, l<256)
// ---------------------------------------------------------------------------
__global__ void prep_sums(const float* __restrict__ class2, const float* __restrict__ wm1,
                          const float* __restrict__ wm2,
                          float* __restrict__ Cs, float* __restrict__ Ws1, float* __restrict__ Ws2)
{
    int t = blockIdx.x * blockDim.x + threadIdx.x;
    if (t < 1536) {
        const float* p = class2 + (size_t)t * 64;
        float s = 0.f;
        for (int l = 0; l < 64; ++l) s += p[l];
        Cs[t] = s;
    } else if (t < 1536 + 2304) {
        int u = t - 1536;
        const float* p = wm1 + (size_t)u * 256;
        float s = 0.f;
        for (int l = 0; l < 256; ++l) s += p[l];
        Ws1[u] = s;
    } else if (t < 1536 + 2304 + 768) {
        int u = t - 3840;
        const float* p = wm2 + (size_t)u * 256;
        float s = 0.f;
        for (int l = 0; l < 256; ++l) s += p[l];
        Ws2[u] = s;
    }
}

// ---------------------------------------------------------------------------
// res1[n,k] = (x2[n,k]*(g.Cs[:,k]) + g.bp1[:,k]) * (r.Ws1[:,k]) + r.bp2[:,k]
// ---------------------------------------------------------------------------
__global__ void res1_kernel(const float* __restrict__ x2, const float* __restrict__ g,
                            const float* __restrict__ r,
                            const float* __restrict__ Cs, const float* __restrict__ bp1,
                            const float* __restrict__ Ws1, const float* __restrict__ bp2,
                            float* __restrict__ res1)
{
    int idx = blockIdx.x * blockDim.x + threadIdx.x;
    if (idx >= N_NODES * 768) return;
    int n = idx / 768, k = idx % 768;
    float g0 = g[n * 2 + 0], g1 = g[n * 2 + 1];
    float a = x2[idx] * (g0 * Cs[k] + g1 * Cs[768 + k]) + g0 * bp1[k] + g1 * bp1[768 + k];
    float r0 = r[n * 3 + 0], r1 = r[n * 3 + 1], r2 = r[n * 3 + 2];
    res1[idx] = a * (r0 * Ws1[k] + r1 * Ws1[768 + k] + r2 * Ws1[1536 + k])
              + r0 * bp2[k] + r1 * bp2[768 + k] + r2 * bp2[1536 + k];
}

// ---------------------------------------------------------------------------
// Graph pieces
// ---------------------------------------------------------------------------
__global__ void deg_count(const int* __restrict__ dst, float* __restrict__ deg)
{
    int e = blockIdx.x * blockDim.x + threadIdx.x;
    if (e < N_EDGES) atomicAdd(&deg[dst[e]], 1.0f);
}

__global__ void dinv_kernel(const float* __restrict__ deg, float* __restrict__ dinv)
{
    int n = blockIdx.x * blockDim.x + threadIdx.x;
    if (n >= N_NODES) return;
    float d = deg[n];
    dinv[n] = (d > 0.f) ? rsqrtf(fmaxf(d, 1.0f)) : 0.0f;
}

// agg[dst] += h[src] * dinv[src]*dinv[dst];  one block per edge, F threads.
__global__ void scatter_kernel(const float* __restrict__ h, const int* __restrict__ src,
                               const int* __restrict__ dst, const float* __restrict__ dinv,
                               float* __restrict__ agg, int F)
{
    int e = blockIdx.x;
    int k = threadIdx.x;
    int s = src[e], d = dst[e];
    float c = dinv[s] * dinv[d];
    atomicAdd(&agg[(size_t)d * F + k], h[(size_t)s * F + k] * c);
}

// ---------------------------------------------------------------------------
// h1 = (r@wm12)*relu(agg1+gcn1_b) + 2e-4*(res1@wm13);  res2 = h1*(r.Ws2[:,k])
// ---------------------------------------------------------------------------
__global__ void combine_kernel(const float* __restrict__ agg1, const float* __restrict__ gcn1_b,
                               const float* __restrict__ t13, const float* __restrict__ r,
                               const float* __restrict__ wm12, const float* __restrict__ Ws2,
                               float* __restrict__ res2)
{
    int idx = blockIdx.x * blockDim.x + threadIdx.x;
    if (idx >= N_NODES * 256) return;
    int n = idx >> 8, k = idx & 255;
    float r0 = r[n * 3 + 0], r1 = r[n * 3 + 1], r2 = r[n * 3 + 2];
    float s12 = r0 * wm12[k] + r1 * wm12[256 + k] + r2 * wm12[512 + k];
    float h1v = fmaxf(agg1[idx] + gcn1_b[k], 0.f);
    float h1  = s12 * h1v + 2e-4f * t13[idx];
    res2[idx] = h1 * (r0 * Ws2[k] + r1 * Ws2[256 + k] + r2 * Ws2[512 + k]);
}

// ---------------------------------------------------------------------------
// out = log_softmax(relu(agg2 + gcn2_b) @ fc_W + fc_b)  -> [N,8]
// ---------------------------------------------------------------------------
__global__ void final_kernel(const float* __restrict__ agg2, const float* __restrict__ gcn2_b,
                             const float* __restrict__ fc_W, const float* __restrict__ fc_b,
                             float* __restrict__ out)
{
    int n = blockIdx.x * blockDim.x + threadIdx.x;
    if (n >= N_NODES) return;
    float z[8];
    #pragma unroll
    for (int j = 0; j < 8; ++j) z[j] = fc_b[j];
    const float* a = agg2 + (size_t)n * 32;
    for (int k = 0; k < 32; ++k) {
        float h = fmaxf(a[k] + gcn2_b[k], 0.f);
        #pragma unroll
        for (int j = 0; j < 8; ++j) z[j] += h * fc_W[k * 8 + j];
    }
    float m = z[0];
    #pragma unroll
    for (int j = 1; j < 8; ++j) m = fmaxf(m, z[j]);
    float s = 0.f;
    #pragma unroll
    for (int j = 0; j < 8; ++j) s += __expf(z[j] - m);
    float ls = __logf(s);
    #pragma unroll
    for (int j = 0; j < 8; ++j) out[n * 8 + j] = z[j] - m - ls;
}

// ---------------------------------------------------------------------------
// Host-side orchestration
// ---------------------------------------------------------------------------
extern "C" void kernel_launch(void* const* d_in, const int* in_sizes, int n_in,
                              void* d_out, int out_size, void* d_ws, size_t ws_size,
                              hipStream_t stream)
{
    const float* x1     = (const float*)d_in[0];
    const float* x11    = (const float*)d_in[1];
    const float* x2     = (const float*)d_in[2];
    const float* mlp_W1 = (const float*)d_in[3];
    const float* mlp_b1 = (const float*)d_in[4];
    const float* mlp_W2 = (const float*)d_in[5];
    const float* mlp_b2 = (const float*)d_in[6];
    const float* mlp_W3 = (const float*)d_in[7];
    const float* mlp_b3 = (const float*)d_in[8];
    const float* m1_W1  = (const float*)d_in[9];
    const float* m1_b1  = (const float*)d_in[10];
    const float* m1_W2  = (const float*)d_in[11];
    const float* m1_b2  = (const float*)d_in[12];
    const float* m1_W3  = (const float*)d_in[13];
    const float* m1_b3  = (const float*)d_in[14];
    const float* wm1    = (const float*)d_in[15];
    const float* bp1    = (const float*)d_in[16];
    const float* bp2    = (const float*)d_in[17];
    const float* wm12   = (const float*)d_in[18];
    const float* wm13   = (const float*)d_in[19];
    const float* class2 = (const float*)d_in[20];
    const float* gcn1_W = (const float*)d_in[21];
    const float* gcn1_b = (const float*)d_in[22];
    const float* wm2    = (const float*)d_in[23];
    const float* gcn2_W = (const float*)d_in[24];
    const float* gcn2_b = (const float*)d_in[25];
    const float* fc_W   = (const float*)d_in[26];
    const float* fc_b   = (const float*)d_in[27];
    const int*   ei     = (const int*)d_in[28];
    const int*   e_src  = ei;
    const int*   e_dst  = ei + N_EDGES;
    float* out = (float*)d_out;

    // ---- workspace layout (floats) ----
    float* ws = (float*)d_ws;
    size_t o = 0;
    float* H1A  = ws + o; o += (size_t)N_NODES * 512;
    float* H2A  = ws + o; o += (size_t)N_NODES * 512;
    float* H1B  = ws + o; o += (size_t)N_NODES * 512;
    float* H2B  = ws + o; o += (size_t)N_NODES * 512;
    float* RES1 = ws + o; o += (size_t)N_NODES * 768;
    float* GH1  = ws + o; o += (size_t)N_NODES * 256;
    float* T13  = ws + o; o += (size_t)N_NODES * 256;
    float* AGG1 = ws + o; o += (size_t)N_NODES * 256;
    float* RES2 = ws + o; o += (size_t)N_NODES * 256;
    float* GH2  = ws + o; o += (size_t)N_NODES * 32;
    float* AGG2 = ws + o; o += (size_t)N_NODES * 32;
    float* Rr   = ws + o; o += (size_t)N_NODES * 3;
    float* Gg   = ws + o; o += (size_t)N_NODES * 2;
    float* CS   = ws + o; o += 2 * 768;
    float* WS1  = ws + o; o += 3 * 768;
    float* WS2  = ws + o; o += 3 * 256;
    float* DEG  = ws + o; o += N_NODES;
    float* DINV = ws + o; o += N_NODES;
    (void)ws_size; (void)n_in; (void)in_sizes; (void)out_size;

    dim3 blk(128);
    // MLP A: relu(x1@W1+b1), relu(.@W2+b2)   (16x64 tiles per wave, NT=4)
    gemm_f32_wmma<4><<<dim3(512/64, 1024/64), blk, 0, stream>>>(x1,  mlp_W1, mlp_b1, H1A, 1024, 512, 768, 3);
    gemm_f32_wmma<4><<<dim3(512/64, 1024/64), blk, 0, stream>>>(H1A, mlp_W2, mlp_b2, H2A, 1024, 512, 512, 3);
    // MLP B: relu(x11@W1+b1), relu(.@W2+b2)
    gemm_f32_wmma<4><<<dim3(512/64, 1024/64), blk, 0, stream>>>(x11, m1_W1,  m1_b1,  H1B, 1024, 512, 768, 3);
    gemm_f32_wmma<4><<<dim3(512/64, 1024/64), blk, 0, stream>>>(H1B, m1_W2,  m1_b2,  H2B, 1024, 512, 512, 3);

    // heads + routing
    route_kernel<<<(N_NODES + 127) / 128, 128, 0, stream>>>(H2A, H2B, mlp_W3, mlp_b3, m1_W3, m1_b3, Rr, Gg);

    // collapse einsum weights over last axis
    prep_sums<<<(4608 + 255) / 256, 256, 0, stream>>>(class2, wm1, wm2, CS, WS1, WS2);

    // res1 (fused einsums)
    res1_kernel<<<(N_NODES * 768 + 255) / 256, 256, 0, stream>>>(x2, Gg, Rr, CS, bp1, WS1, bp2, RES1);

    // dense GEMMs feeding GCN1 path
    gemm_f32_wmma<4><<<dim3(256/64, 1024/64), blk, 0, stream>>>(RES1, gcn1_W, nullptr, GH1, 1024, 256, 768, 0);
    gemm_f32_wmma<4><<<dim3(256/64, 1024/64), blk, 0, stream>>>(RES1, wm13,   nullptr, T13, 1024, 256, 768, 0);

    // degrees
    hipMemsetAsync(DEG, 0, N_NODES * sizeof(float), stream);
    deg_count<<<(N_EDGES + 255) / 256, 256, 0, stream>>>(e_dst, DEG);
    dinv_kernel<<<(N_NODES + 255) / 256, 256, 0, stream>>>(DEG, DINV);

    // GCN1 aggregate
    hipMemsetAsync(AGG1, 0, (size_t)N_NODES * 256 * sizeof(float), stream);
    scatter_kernel<<<N_EDGES, 256, 0, stream>>>(GH1, e_src, e_dst, DINV, AGG1, 256);

    // h1 gating + res2
    combine_kernel<<<(N_NODES * 256 + 255) / 256, 256, 0, stream>>>(AGG1, gcn1_b, T13, Rr, wm12, WS2, RES2);

    // GCN2 (N=32 -> NT=2, one 32-wide strip)
    gemm_f32_wmma<2><<<dim3(32/32, 1024/64), blk, 0, stream>>>(RES2, gcn2_W, nullptr, GH2, 1024, 32, 256, 0);
    hipMemsetAsync(AGG2, 0, (size_t)N_NODES * 32 * sizeof(float), stream);
    scatter_kernel<<<N_EDGES, 32, 0, stream>>>(GH2, e_src, e_dst, DINV, AGG2, 32);

    // classifier + log_softmax
    final_kernel<<<(N_NODES + 255) / 256, 256, 0, stream>>>(AGG2, gcn2_b, fc_W, fc_b, out);
}